// MambaConv1DTest_79182017069423
// MI455X (gfx1250) — compile-verified
//
#include <hip/hip_runtime.h>
#include <math.h>

// Problem dims (fixed by reference)
#define B_  32
#define C_  1024
#define L_  4096
#define K_  4
#define NC_ 10

#define QMAXF 127.0f
#define QMINF -128.0f

// Workspace layout (floats)
#define WS_AMAX_X    0
#define WS_AMAX_H1   1
#define WS_AMAX_S    2
#define WS_AMAX_POOL 3
#define WS_AMAX_LW   4
#define WS_WQ        64                  // fq'd conv weights, C_*K_ floats
#define WS_POOLED    (64 + C_*K_)        // pooled activations, B_*C_ floats
#define WS_TOTAL     (WS_POOLED + B_*C_)

typedef __attribute__((ext_vector_type(2))) float v2f;
typedef __attribute__((ext_vector_type(8))) float v8f;

__device__ __forceinline__ float fq_apply(float v, float s, float inv_s) {
  // quantize-dequantize: clip(rne_round(v/s), -128, 127) * s
  float q = rintf(v * inv_s);
  q = fminf(fmaxf(q, QMINF), QMAXF);
  return q * s;
}

__device__ __forceinline__ float scale_of(float amax) {
  return amax * (1.0f / QMAXF) + 1e-8f;
}

// ---------------- init workspace ----------------
__global__ void k_init_ws(float* __restrict__ ws) {
  int i = blockIdx.x * blockDim.x + threadIdx.x;
  if (i < WS_TOTAL) ws[i] = 0.0f;
}

// ---------------- pass 1: abs-max over x ----------------
__global__ void k_absmax_x(const float4* __restrict__ x4, float* __restrict__ ws, int n4) {
  float m = 0.0f;
  for (int i = blockIdx.x * blockDim.x + threadIdx.x; i < n4; i += gridDim.x * blockDim.x) {
    float4 v = x4[i];
    m = fmaxf(m, fmaxf(fmaxf(fabsf(v.x), fabsf(v.y)), fmaxf(fabsf(v.z), fabsf(v.w))));
  }
  __shared__ float red[256];
  int tid = threadIdx.x;
  red[tid] = m;
  __syncthreads();
  for (int o = 128; o > 0; o >>= 1) {
    if (tid < o) red[tid] = fmaxf(red[tid], red[tid + o]);
    __syncthreads();
  }
  if (tid == 0)
    atomicMax(reinterpret_cast<unsigned int*>(ws) + WS_AMAX_X, __float_as_uint(red[0]));
}

// ---------------- weight prep: fq(conv_w) + abs-max(lin_w) ----------------
__global__ void k_prep_weights(const float* __restrict__ conv_w,
                               const float* __restrict__ lin_w,
                               float* __restrict__ ws) {
  __shared__ float red[256];
  const int tid = threadIdx.x;

  // abs-max over conv_w (C_*K_ = 4096)
  float m = 0.0f;
  for (int i = tid; i < C_ * K_; i += 256) m = fmaxf(m, fabsf(conv_w[i]));
  red[tid] = m;
  __syncthreads();
  for (int o = 128; o > 0; o >>= 1) {
    if (tid < o) red[tid] = fmaxf(red[tid], red[tid + o]);
    __syncthreads();
  }
  const float amax_cw = red[0];
  __syncthreads();

  // abs-max over lin_w (NC_*C_ = 10240)
  float m2 = 0.0f;
  for (int i = tid; i < NC_ * C_; i += 256) m2 = fmaxf(m2, fabsf(lin_w[i]));
  red[tid] = m2;
  __syncthreads();
  for (int o = 128; o > 0; o >>= 1) {
    if (tid < o) red[tid] = fmaxf(red[tid], red[tid + o]);
    __syncthreads();
  }
  if (tid == 0) ws[WS_AMAX_LW] = red[0];

  // write fake-quantized conv weights
  const float s = scale_of(amax_cw);
  const float inv_s = 1.0f / s;
  for (int i = tid; i < C_ * K_; i += 256) ws[WS_WQ + i] = fq_apply(conv_w[i], s, inv_s);
}

// ---------------- streaming passes over x (recompute conv chain) ----------------
// PASS==2: reduce abs-max of conv output h1
// PASS==3: reduce abs-max of silu(fq(h1))
// PASS==4: full chain -> mean over L -> pooled[row]
template <int PASS>
__global__ void k_stream(const float* __restrict__ x,
                         const float* __restrict__ conv_b,
                         float* __restrict__ ws) {
  const int tid = threadIdx.x;           // 256 threads
  const int row = blockIdx.x;            // b*C + c, one row of length L
  const int c = row & (C_ - 1);
  const float* __restrict__ xr = x + (size_t)row * L_;

  const float s_x = scale_of(ws[WS_AMAX_X]);
  const float inv_x = 1.0f / s_x;
  const float w0 = ws[WS_WQ + c * 4 + 0];
  const float w1 = ws[WS_WQ + c * 4 + 1];
  const float w2 = ws[WS_WQ + c * 4 + 2];
  const float w3 = ws[WS_WQ + c * 4 + 3];
  const float bias = conv_b[c];

  float s_c = 1.0f, inv_c = 1.0f, s_s = 1.0f, inv_sv = 1.0f, s_h3 = 1.0f, inv_h3 = 1.0f;
  if (PASS >= 3) {
    s_c = scale_of(ws[WS_AMAX_H1]);
    inv_c = 1.0f / s_c;
  }
  if (PASS >= 4) {
    const float amax_s = ws[WS_AMAX_S];
    s_s = scale_of(amax_s);
    inv_sv = 1.0f / s_s;
    // |fq(y)| is monotone in |y|  =>  max|fq(s)| = fq(max|s|): no extra pass needed
    const float amax_h3 = fq_apply(amax_s, s_s, inv_sv);
    s_h3 = scale_of(amax_h3);
    inv_h3 = 1.0f / s_h3;
  }

  __shared__ float sx[1024 + 3];
  __shared__ float red[256];
  float acc = 0.0f;  // running max (pass 2/3) or sum (pass 4)

  for (int t = 0; t < 4; ++t) {
    if (t < 3) {  // prefetch next L-tile -> global_prefetch_b8
      __builtin_prefetch(xr + (t + 1) * 1024 + tid * 4, 0, 1);
    }
    // stage fq(x) tile (+3 causal halo) into LDS
    for (int i = tid; i < 1024 + 3; i += 256) {
      const int l = t * 1024 + i - 3;
      const float v = (l >= 0) ? xr[l] : 0.0f;
      sx[i] = fq_apply(v, s_x, inv_x);
    }
    __syncthreads();
    for (int j = 0; j < 4; ++j) {
      const int i = tid + j * 256;
      const float h1 = w0 * sx[i] + w1 * sx[i + 1] + w2 * sx[i + 2] + w3 * sx[i + 3] + bias;
      if (PASS == 2) {
        acc = fmaxf(acc, fabsf(h1));
      } else {
        const float h2 = fq_apply(h1, s_c, inv_c);
        const float sv = h2 / (1.0f + __expf(-h2));  // h2 * sigmoid(h2)
        if (PASS == 3) {
          acc = fmaxf(acc, fabsf(sv));
        } else {
          const float h3 = fq_apply(sv, s_s, inv_sv);
          const float h4 = fq_apply(h3, s_h3, inv_h3);
          acc += h4;
        }
      }
    }
    __syncthreads();
  }

  red[tid] = acc;
  __syncthreads();
  for (int o = 128; o > 0; o >>= 1) {
    if (tid < o) red[tid] = (PASS == 4) ? (red[tid] + red[tid + o]) : fmaxf(red[tid], red[tid + o]);
    __syncthreads();
  }
  if (tid == 0) {
    if (PASS == 2)
      atomicMax(reinterpret_cast<unsigned int*>(ws) + WS_AMAX_H1, __float_as_uint(red[0]));
    else if (PASS == 3)
      atomicMax(reinterpret_cast<unsigned int*>(ws) + WS_AMAX_S, __float_as_uint(red[0]));
    else
      ws[WS_POOLED + row] = red[0] * (1.0f / (float)L_);
  }
}

// ---------------- abs-max over pooled [B,C] ----------------
__global__ void k_absmax_pooled(float* __restrict__ ws) {
  const int tid = threadIdx.x;  // 1024
  float m = 0.0f;
  for (int i = tid; i < B_ * C_; i += 1024) m = fmaxf(m, fabsf(ws[WS_POOLED + i]));
  __shared__ float red[1024];
  red[tid] = m;
  __syncthreads();
  for (int o = 512; o > 0; o >>= 1) {
    if (tid < o) red[tid] = fmaxf(red[tid], red[tid + o]);
    __syncthreads();
  }
  if (tid == 0) ws[WS_AMAX_POOL] = red[0];
}

// ---------------- final linear via V_WMMA_F32_16X16X4_F32 ----------------
// logits[32,10] = fq(pooled) @ fq(lin_w)^T + lin_b ; one wave32, two M-tiles.
// A 16x4 f32 frag: lane<16 holds K={0,1}, lane>=16 holds K={2,3}, row M = lane&15.
// B 4x16 frag: mirrored (K from lane-half, N = lane&15). D 16x16: VGPR v -> M=v+8*(lane/16), N=lane&15.
__global__ void k_gemm_wmma(const float* __restrict__ lin_w,
                            const float* __restrict__ lin_b,
                            const float* __restrict__ ws,
                            float* __restrict__ out) {
  const int lane = threadIdx.x;  // 32
  const int half = lane >> 4;
  const int ln = lane & 15;
  const float* __restrict__ pooled = ws + WS_POOLED;

  const float s_p = scale_of(ws[WS_AMAX_POOL]);
  const float ip = 1.0f / s_p;
  const float s_w = scale_of(ws[WS_AMAX_LW]);
  const float iw = 1.0f / s_w;

  v8f acc0 = {};  // rows 0..15
  v8f acc1 = {};  // rows 16..31
  const int rA0 = ln;
  const int rA1 = 16 + ln;

  for (int k0 = 0; k0 < C_; k0 += 4) {
    const int k = k0 + half * 2;
    v2f a0, a1, b;
    a0[0] = fq_apply(pooled[rA0 * C_ + k + 0], s_p, ip);
    a0[1] = fq_apply(pooled[rA0 * C_ + k + 1], s_p, ip);
    a1[0] = fq_apply(pooled[rA1 * C_ + k + 0], s_p, ip);
    a1[1] = fq_apply(pooled[rA1 * C_ + k + 1], s_p, ip);
    if (ln < NC_) {
      b[0] = fq_apply(lin_w[ln * C_ + k + 0], s_w, iw);
      b[1] = fq_apply(lin_w[ln * C_ + k + 1], s_w, iw);
    } else {
      b[0] = 0.0f;
      b[1] = 0.0f;
    }
    acc0 = __builtin_amdgcn_wmma_f32_16x16x4_f32(false, a0, false, b, (short)0, acc0, false, false);
    acc1 = __builtin_amdgcn_wmma_f32_16x16x4_f32(false, a1, false, b, (short)0, acc1, false, false);
  }

  if (ln < NC_) {
    const float bb = lin_b[ln];
    for (int v = 0; v < 8; ++v) {
      const int m = v + half * 8;
      out[m * NC_ + ln] = acc0[v] + bb;
      out[(m + 16) * NC_ + ln] = acc1[v] + bb;
    }
  }
}

extern "C" void kernel_launch(void* const* d_in, const int* in_sizes, int n_in,
                              void* d_out, int out_size, void* d_ws, size_t ws_size,
                              hipStream_t stream) {
  (void)in_sizes; (void)n_in; (void)out_size; (void)ws_size;
  const float* x      = (const float*)d_in[0];
  const float* conv_w = (const float*)d_in[1];
  const float* conv_b = (const float*)d_in[2];
  const float* lin_w  = (const float*)d_in[3];
  const float* lin_b  = (const float*)d_in[4];
  float* out = (float*)d_out;
  float* ws  = (float*)d_ws;

  k_init_ws<<<(WS_TOTAL + 255) / 256, 256, 0, stream>>>(ws);

  const int n4 = (B_ * C_ * L_) / 4;
  k_absmax_x<<<2048, 256, 0, stream>>>((const float4*)x, ws, n4);

  k_prep_weights<<<1, 256, 0, stream>>>(conv_w, lin_w, ws);

  k_stream<2><<<B_ * C_, 256, 0, stream>>>(x, conv_b, ws);
  k_stream<3><<<B_ * C_, 256, 0, stream>>>(x, conv_b, ws);
  k_stream<4><<<B_ * C_, 256, 0, stream>>>(x, conv_b, ws);

  k_absmax_pooled<<<1, 1024, 0, stream>>>(ws);

  k_gemm_wmma<<<1, 32, 0, stream>>>(lin_w, lin_b, ws, out);
}